// SampleNetDC_73555609911851
// MI455X (gfx1250) — compile-verified
//
#include <hip/hip_runtime.h>
#include <hip/hip_bf16.h>
#include <math.h>

// ---- problem constants (match reference) ----
#define BB 32
#define PP 2048
#define NN (BB * PP)          // 65536
#define KNN 16
#define LL 8
#define FNR 64
#define KS 5
#define KS3 (KS * KS * KS)    // 125
#define NC 40
#define EPS 1e-5f

typedef __attribute__((ext_vector_type(2))) float v2f;
typedef __attribute__((ext_vector_type(8))) float v8f;

__device__ __forceinline__ v8f wmma_f32_16x16x4(v2f a, v2f b, v8f c) {
  // D = A(16x4) * B(4x16) + C ; 8 args: neg_a, A, neg_b, B, c_mod, C, reuse_a, reuse_b
  return __builtin_amdgcn_wmma_f32_16x16x4_f32(false, a, false, b, (short)0, c, false, false);
}

// =====================================================================
// Kernel 1: per-cloud KNN via WMMA Gram tiles.
// grid = BB*16 blocks, 128 threads (4 waves). Wave owns 32 rows (2 WMMA tiles).
// All 2048 cloud positions + squared norms staged in LDS.
// =====================================================================
__global__ __launch_bounds__(128) void knn_kernel(const float* __restrict__ pos,
                                                  int* __restrict__ nbr) {
  __shared__ float sx[PP], sy[PP], sz[PP], ssq[PP];
  __shared__ float s_dist[4][32 * 16];

  const int batch = blockIdx.x >> 4;
  const int chunk = blockIdx.x & 15;
  const int tid = threadIdx.x;

  // stage the whole cloud into LDS
  for (int j = tid; j < PP; j += 128) {
    const float* p = pos + (size_t)(batch * PP + j) * 3;
    __builtin_prefetch(p + 96, 0, 0);   // global_prefetch_b8 path
    float x = p[0], y = p[1], z = p[2];
    sx[j] = x; sy[j] = y; sz[j] = z;
    ssq[j] = x * x + y * y + z * z;
  }
  __syncthreads();

  const int lane = tid & 31;
  const int wave = tid >> 5;
  const bool hi = lane >= 16;
  const int nsel = lane & 15;
  const int rowchunk = chunk * 128 + wave * 32;   // wave's 32 rows (local to cloud)

  // A fragments for the two 16-row tiles (ISA A layout: M=lane&15, K={0,1} lo / {2,3} hi)
  int ar0 = rowchunk + nsel;
  int ar1 = rowchunk + 16 + nsel;
  v2f a0, a1;
  a0[0] = hi ? sz[ar0] : sx[ar0];
  a0[1] = hi ? 0.0f    : sy[ar0];
  a1[0] = hi ? sz[ar1] : sx[ar1];
  a1[1] = hi ? 0.0f    : sy[ar1];

  // squared norms for rows produced by this lane's 8 accum VGPRs (M = r + 8*hi)
  float sr0[8], sr1[8];
#pragma unroll
  for (int r = 0; r < 8; ++r) {
    sr0[r] = ssq[rowchunk + r + (hi ? 8 : 0)];
    sr1[r] = ssq[rowchunk + 16 + r + (hi ? 8 : 0)];
  }

  // top-16 (sorted ascending) per owned row (lane owns row rowchunk+lane)
  float td[KNN]; int ti[KNN];
#pragma unroll
  for (int t = 0; t < KNN; ++t) { td[t] = 1e30f; ti[t] = 0; }

  for (int ct = 0; ct < PP / 16; ++ct) {
    const int colbase = ct * 16;
    const int c = colbase + nsel;
    v2f bf;
    bf[0] = hi ? sz[c] : sx[c];
    bf[1] = hi ? 0.0f  : sy[c];
    const float sqc = ssq[c];

    v8f zc = {0.f, 0.f, 0.f, 0.f, 0.f, 0.f, 0.f, 0.f};
    v8f g0 = wmma_f32_16x16x4(a0, bf, zc);
    v8f g1 = wmma_f32_16x16x4(a1, bf, zc);

#pragma unroll
    for (int r = 0; r < 8; ++r) {
      const int m = r + (hi ? 8 : 0);
      float d0 = sr0[r] + sqc - 2.0f * g0[r];
      float d1 = sr1[r] + sqc - 2.0f * g1[r];
      if (rowchunk + m == c)      d0 = 1e30f;   // mask self
      if (rowchunk + 16 + m == c) d1 = 1e30f;
      s_dist[wave][m * 16 + nsel]        = d0;
      s_dist[wave][(16 + m) * 16 + nsel] = d1;
    }
    __builtin_amdgcn_wave_barrier();   // same-wave DS ops are in-order; block reordering

    // each lane scans the 16 new candidates of its own row
#pragma unroll
    for (int j = 0; j < 16; ++j) {
      float d = s_dist[wave][lane * 16 + j];
      int gidx = colbase + j;
      if (d < td[KNN - 1]) {
        float cd = d; int ci = gidx;
#pragma unroll
        for (int t = 0; t < KNN; ++t) {
          if (cd < td[t]) {
            float fd = td[t]; td[t] = cd; cd = fd;
            int   fi = ti[t]; ti[t] = ci; ci = fi;
          }
        }
      }
    }
    __builtin_amdgcn_wave_barrier();
  }

  const int gi = batch * PP + rowchunk + lane;
#pragma unroll
  for (int t = 0; t < KNN; ++t) nbr[gi * KNN + t] = batch * PP + ti[t];
}

// =====================================================================
// Kernel 2: per-point local frame (5 QR iters on 3x3 cov) + trilinear
// spline feature accumulation. One thread per point; spline table in LDS.
// =====================================================================
__global__ __launch_bounds__(256) void feature_kernel(const float* __restrict__ pos,
                                                      const int* __restrict__ nbr,
                                                      const float* __restrict__ w_spline,
                                                      const float* __restrict__ w_root,
                                                      const float* __restrict__ b_spline,
                                                      float* __restrict__ ond) {
  __shared__ __align__(16) float s_spl[KS3 * FNR];   // 8000 floats = 32KB
  __shared__ float s_root[FNR], s_bsp[FNR];
  for (int j = threadIdx.x; j < KS3 * FNR; j += 256) s_spl[j] = w_spline[j];
  if (threadIdx.x < FNR) {
    s_root[threadIdx.x] = w_root[threadIdx.x];
    s_bsp[threadIdx.x]  = b_spline[threadIdx.x];
  }
  __syncthreads();

  const int i = blockIdx.x * 256 + threadIdx.x;
  const float px = pos[3 * i], py = pos[3 * i + 1], pz = pos[3 * i + 2];

  float cl[KNN][3];
#pragma unroll 1
  for (int k = 0; k < KNN; ++k) {
    int nb = nbr[i * KNN + k];
    cl[k][0] = pos[3 * nb]     - px;
    cl[k][1] = pos[3 * nb + 1] - py;
    cl[k][2] = pos[3 * nb + 2] - pz;
  }

  // covariance of first LL neighbors
  float A[3][3] = {{0,0,0},{0,0,0},{0,0,0}};
#pragma unroll
  for (int k = 0; k < LL; ++k) {
    A[0][0] += cl[k][0] * cl[k][0];
    A[0][1] += cl[k][0] * cl[k][1];
    A[0][2] += cl[k][0] * cl[k][2];
    A[1][1] += cl[k][1] * cl[k][1];
    A[1][2] += cl[k][1] * cl[k][2];
    A[2][2] += cl[k][2] * cl[k][2];
  }
  A[1][0] = A[0][1]; A[2][0] = A[0][2]; A[2][1] = A[1][2];

  float Vt[3][3] = {{1,0,0},{0,1,0},{0,0,1}};
#pragma unroll 1
  for (int it = 0; it < 5; ++it) {
    // modified Gram-Schmidt QR on columns of A
    float c0x = A[0][0], c0y = A[1][0], c0z = A[2][0];
    float c1x = A[0][1], c1y = A[1][1], c1z = A[2][1];
    float c2x = A[0][2], c2y = A[1][2], c2z = A[2][2];
    float r00 = sqrtf(c0x*c0x + c0y*c0y + c0z*c0z + 1e-30f);
    float in0 = 1.0f / r00;
    float q0x = c0x*in0, q0y = c0y*in0, q0z = c0z*in0;
    float r01 = q0x*c1x + q0y*c1y + q0z*c1z;
    c1x -= r01*q0x; c1y -= r01*q0y; c1z -= r01*q0z;
    float r11 = sqrtf(c1x*c1x + c1y*c1y + c1z*c1z + 1e-30f);
    float in1 = 1.0f / r11;
    float q1x = c1x*in1, q1y = c1y*in1, q1z = c1z*in1;
    float r02 = q0x*c2x + q0y*c2y + q0z*c2z;
    c2x -= r02*q0x; c2y -= r02*q0y; c2z -= r02*q0z;
    float r12 = q1x*c2x + q1y*c2y + q1z*c2z;
    c2x -= r12*q1x; c2y -= r12*q1y; c2z -= r12*q1z;
    float r22 = sqrtf(c2x*c2x + c2y*c2y + c2z*c2z + 1e-30f);
    float in2 = 1.0f / r22;
    float q2x = c2x*in2, q2y = c2y*in2, q2z = c2z*in2;

    float Q[3][3] = {{q0x,q1x,q2x},{q0y,q1y,q2y},{q0z,q1z,q2z}};
    float NA[3][3];
#pragma unroll
    for (int j = 0; j < 3; ++j) {
      NA[0][j] = r00*Q[0][j] + r01*Q[1][j] + r02*Q[2][j];
      NA[1][j] =               r11*Q[1][j] + r12*Q[2][j];
      NA[2][j] =                             r22*Q[2][j];
    }
    float NV[3][3];
#pragma unroll
    for (int r = 0; r < 3; ++r)
#pragma unroll
      for (int j = 0; j < 3; ++j)
        NV[r][j] = Vt[r][0]*Q[0][j] + Vt[r][1]*Q[1][j] + Vt[r][2]*Q[2][j];
#pragma unroll
    for (int r = 0; r < 3; ++r)
#pragma unroll
      for (int j = 0; j < 3; ++j) { A[r][j] = NA[r][j]; Vt[r][j] = NV[r][j]; }
  }

  // rotate clusters in place: dc = cl @ Vt
#pragma unroll
  for (int k = 0; k < KNN; ++k) {
    float x = cl[k][0], y = cl[k][1], z = cl[k][2];
    cl[k][0] = x*Vt[0][0] + y*Vt[1][0] + z*Vt[2][0];
    cl[k][1] = x*Vt[0][1] + y*Vt[1][1] + z*Vt[2][1];
    cl[k][2] = x*Vt[0][2] + y*Vt[1][2] + z*Vt[2][2];
  }
  float ssum = 0.0f;
#pragma unroll
  for (int k = 0; k < KNN; ++k) ssum += cl[k][2];
  float sgn = (ssum > 0.0f) ? 1.0f : ((ssum < 0.0f) ? -1.0f : 0.0f);
#pragma unroll
  for (int k = 0; k < KNN; ++k) cl[k][2] *= sgn;

  float ma = 0.0f;
#pragma unroll
  for (int k = 0; k < KNN; ++k)
#pragma unroll
    for (int d = 0; d < 3; ++d) ma = fmaxf(ma, fabsf(cl[k][d]));
  float inv = 0.5f / ma;

  float acc[FNR];
#pragma unroll
  for (int q = 0; q < FNR; ++q) acc[q] = 0.0f;

#pragma unroll 1
  for (int k = 0; k < KNN; ++k) {
    int   i0[3]; float fr[3];
#pragma unroll
    for (int d = 0; d < 3; ++d) {
      float pv = (cl[k][d] * inv + 0.5f) * (float)(KS - 1);
      int ii = (int)floorf(pv);
      ii = ii < 0 ? 0 : (ii > KS - 2 ? KS - 2 : ii);
      i0[d] = ii; fr[d] = pv - (float)ii;
    }
#pragma unroll 1
    for (int c = 0; c < 8; ++c) {
      int b0 = c & 1, b1 = (c >> 1) & 1, b2 = (c >> 2) & 1;
      int kidx = (i0[0] + b0) + (i0[1] + b1) * KS + (i0[2] + b2) * KS * KS;
      float w = (b0 ? fr[0] : 1.0f - fr[0]) *
                (b1 ? fr[1] : 1.0f - fr[1]) *
                (b2 ? fr[2] : 1.0f - fr[2]);
      const float4* sp = reinterpret_cast<const float4*>(&s_spl[kidx * FNR]);
#pragma unroll
      for (int q = 0; q < FNR / 4; ++q) {
        float4 s = sp[q];
        acc[4*q+0] += w * s.x; acc[4*q+1] += w * s.y;
        acc[4*q+2] += w * s.z; acc[4*q+3] += w * s.w;
      }
    }
  }

  const float inv16 = 1.0f / (float)KNN;
  float4* op = reinterpret_cast<float4*>(ond + (size_t)i * FNR);
#pragma unroll
  for (int q = 0; q < FNR / 4; ++q) {
    float4 o;
    o.x = acc[4*q+0]*inv16 + s_root[4*q+0] + s_bsp[4*q+0];
    o.y = acc[4*q+1]*inv16 + s_root[4*q+1] + s_bsp[4*q+1];
    o.z = acc[4*q+2]*inv16 + s_root[4*q+2] + s_bsp[4*q+2];
    o.w = acc[4*q+3]*inv16 + s_root[4*q+3] + s_bsp[4*q+3];
    op[q] = o;
  }
}

// =====================================================================
// Kernel 3a/3b: batchnorm statistics (sum, sumsq per feature)
// =====================================================================
__global__ void zero_stats_kernel(float* __restrict__ stats) {
  if (threadIdx.x < 2 * FNR) stats[threadIdx.x] = 0.0f;
}

__global__ __launch_bounds__(256) void bn_stats_kernel(const float* __restrict__ ond,
                                                       float* __restrict__ stats) {
  const int f = threadIdx.x & 63;
  const int g = threadIdx.x >> 6;        // 0..3
  const int rowbase = blockIdx.x * 256;  // 256 blocks cover N rows
  float s = 0.0f, s2 = 0.0f;
  for (int r = g; r < 256; r += 4) {
    float x = ond[(size_t)(rowbase + r) * FNR + f];
    s += x; s2 += x * x;
  }
  __shared__ float red[256], red2[256];
  red[threadIdx.x] = s; red2[threadIdx.x] = s2;
  __syncthreads();
  if (g == 0) {
#pragma unroll
    for (int gg = 1; gg < 4; ++gg) { s += red[f + 64 * gg]; s2 += red2[f + 64 * gg]; }
    atomicAdd(&stats[f], s);
    atomicAdd(&stats[FNR + f], s2);
  }
}

// =====================================================================
// Kernel 4: BN + sigmoid + per-cloud mean -> ys (32 x 64)
// =====================================================================
__global__ __launch_bounds__(256) void bn_apply_kernel(const float* __restrict__ ond,
                                                       const float* __restrict__ stats,
                                                       const float* __restrict__ gamma,
                                                       const float* __restrict__ beta,
                                                       float* __restrict__ ys) {
  const int b = blockIdx.x;
  const int f = threadIdx.x & 63;
  const int g = threadIdx.x >> 6;
  const float invN = 1.0f / (float)NN;
  float mu  = stats[f] * invN;
  float var = stats[FNR + f] * invN - mu * mu;
  float sc  = gamma[f] * rsqrtf(var + EPS);
  float bb  = beta[f];
  float s = 0.0f;
  for (int p = g; p < PP; p += 4) {
    float x = ond[(size_t)(b * PP + p) * FNR + f];
    float y = (x - mu) * sc + bb;
    s += 1.0f / (1.0f + expf(-y));
  }
  __shared__ float red[256];
  red[threadIdx.x] = s;
  __syncthreads();
  if (g == 0) {
#pragma unroll
    for (int gg = 1; gg < 4; ++gg) s += red[f + 64 * gg];
    ys[b * FNR + f] = s * (1.0f / (float)PP);
  }
}

// =====================================================================
// Kernel 5: ys(32x64) @ w1(64x256) + b1 -> elu -> y1 via WMMA f32 16x16x4
// one block, 8 waves; wave handles (mtile, ntile) pairs; 16 K-steps each.
// =====================================================================
__global__ __launch_bounds__(256) void mlp1_kernel(const float* __restrict__ ys,
                                                   const float* __restrict__ w1,
                                                   const float* __restrict__ b1,
                                                   float* __restrict__ y1) {
  const int lane = threadIdx.x & 31;
  const int wave = threadIdx.x >> 5;   // 0..7
  const bool hi = lane >= 16;
  const int nsel = lane & 15;

  for (int t = wave; t < 32; t += 8) {   // 2 mtiles x 16 ntiles
    const int mt = t & 1;
    const int nt = t >> 1;
    v8f c = {0.f, 0.f, 0.f, 0.f, 0.f, 0.f, 0.f, 0.f};
    const int arow = mt * 16 + nsel;
    const int bcol = nt * 16 + nsel;
#pragma unroll
    for (int k4 = 0; k4 < 16; ++k4) {
      const int k0 = k4 * 4;
      v2f a, bb;
      a[0] = ys[arow * FNR + k0 + (hi ? 2 : 0)];
      a[1] = ys[arow * FNR + k0 + (hi ? 3 : 1)];
      bb[0] = w1[(k0 + (hi ? 2 : 0)) * 256 + bcol];
      bb[1] = w1[(k0 + (hi ? 3 : 1)) * 256 + bcol];
      c = wmma_f32_16x16x4(a, bb, c);
    }
#pragma unroll
    for (int r = 0; r < 8; ++r) {
      const int row = mt * 16 + r + (hi ? 8 : 0);
      const int col = nt * 16 + nsel;
      float v = c[r] + b1[col];
      v = v > 0.0f ? v : (expf(v) - 1.0f);   // ELU
      y1[row * 256 + col] = v;
    }
  }
}

// =====================================================================
// Kernel 6: y1(32x256) @ w2(256x40) + b2 -> log_softmax -> out (32x40)
// =====================================================================
__global__ __launch_bounds__(256) void mlp2_kernel(const float* __restrict__ y1,
                                                   const float* __restrict__ w2,
                                                   const float* __restrict__ b2,
                                                   float* __restrict__ out) {
  __shared__ float s_log[BB * NC];
  for (int o = threadIdx.x; o < BB * NC; o += 256) {
    const int row = o / NC, col = o % NC;
    float s = b2[col];
    for (int k = 0; k < 256; ++k) s += y1[row * 256 + k] * w2[k * NC + col];
    s_log[o] = s;
  }
  __syncthreads();
  if (threadIdx.x < BB) {
    const int row = threadIdx.x;
    float mx = -3.4e38f;
    for (int c = 0; c < NC; ++c) mx = fmaxf(mx, s_log[row * NC + c]);
    float se = 0.0f;
    for (int c = 0; c < NC; ++c) se += expf(s_log[row * NC + c] - mx);
    float lse = mx + logf(se);
    for (int c = 0; c < NC; ++c) out[row * NC + c] = s_log[row * NC + c] - lse;
  }
}

// =====================================================================
extern "C" void kernel_launch(void* const* d_in, const int* in_sizes, int n_in,
                              void* d_out, int out_size, void* d_ws, size_t ws_size,
                              hipStream_t stream) {
  const float* pos      = (const float*)d_in[0];
  // d_in[1] = batch (unused; derived arithmetically)
  const float* w_spline = (const float*)d_in[2];
  const float* w_root   = (const float*)d_in[3];
  const float* b_spline = (const float*)d_in[4];
  const float* bn_gamma = (const float*)d_in[5];
  const float* bn_beta  = (const float*)d_in[6];
  const float* w1       = (const float*)d_in[7];
  const float* b1       = (const float*)d_in[8];
  const float* w2       = (const float*)d_in[9];
  const float* b2       = (const float*)d_in[10];
  float* out = (float*)d_out;

  char* ws = (char*)d_ws;
  int*   d_nbr   = (int*)(ws);                                   //  4 MiB
  float* d_ond   = (float*)(ws + (size_t)4  * 1024 * 1024);      // 16 MiB
  float* d_stats = (float*)(ws + (size_t)20 * 1024 * 1024);      // 512 B
  float* d_ys    = (float*)(ws + (size_t)20 * 1024 * 1024 + 4096);
  float* d_y1    = (float*)(ws + (size_t)20 * 1024 * 1024 + 16384);

  knn_kernel<<<dim3(BB * 16), dim3(128), 0, stream>>>(pos, d_nbr);
  feature_kernel<<<dim3(NN / 256), dim3(256), 0, stream>>>(pos, d_nbr, w_spline,
                                                           w_root, b_spline, d_ond);
  zero_stats_kernel<<<dim3(1), dim3(128), 0, stream>>>(d_stats);
  bn_stats_kernel<<<dim3(NN / 256), dim3(256), 0, stream>>>(d_ond, d_stats);
  bn_apply_kernel<<<dim3(BB), dim3(256), 0, stream>>>(d_ond, d_stats, bn_gamma,
                                                      bn_beta, d_ys);
  mlp1_kernel<<<dim3(1), dim3(256), 0, stream>>>(d_ys, w1, b1, d_y1);
  mlp2_kernel<<<dim3(1), dim3(256), 0, stream>>>(d_y1, w2, b2, out);
  (void)in_sizes; (void)n_in; (void)out_size; (void)ws_size;
}